// FrequencyRepulsion_40604620816577
// MI455X (gfx1250) — compile-verified
//
#include <hip/hip_runtime.h>

// Problem constants from the reference
#define THR_X 16.0f
#define THR_Y 16.0f
#define FORCE_RATIO 0.5f
#define EPS 0.01f

typedef __attribute__((ext_vector_type(2))) float v2f;
typedef __attribute__((ext_vector_type(8))) float v8f;
typedef __attribute__((ext_vector_type(4))) int   i4v;   // for nontemporal b128 loads

// ---------------------------------------------------------------------------
// Kernel 1: build interleaved node centers, zero force accumulator + output.
// ws layout: ctr = float2[n] (cx,cy), frc = float2[n] (fx,fy) -> 16*n bytes.
// ---------------------------------------------------------------------------
__global__ void repulsion_init(const float* __restrict__ pos,
                               const float* __restrict__ nsx,
                               const float* __restrict__ nsy,
                               float2* __restrict__ ctr,
                               float2* __restrict__ frc,
                               float* __restrict__ out, int n) {
  int i = blockIdx.x * blockDim.x + threadIdx.x;
  if (i == 0) out[0] = 0.0f;
  if (i < n) {
    float2 c;
    c.x = pos[i]     + 0.5f * nsx[i];
    c.y = pos[n + i] + 0.5f * nsy[i];
    ctr[i] = c;
    frc[i] = make_float2(0.0f, 0.0f);
  }
}

// ---------------------------------------------------------------------------
// Kernel 2: pair interaction. 4 pairs per thread; the once-streamed index
// arrays are loaded non-temporally (TH=NT) so the 134MB stream does not
// evict the gather-hot 16MB center array from the 192MB L2. Atomics are
// branched out for the ~99.9% of pairs outside the threshold window.
// ---------------------------------------------------------------------------
__device__ __forceinline__ void do_pair(int s, int d,
                                        const float2* __restrict__ ctr,
                                        float2* __restrict__ frc) {
  float2 cs = ctr[s];
  float2 cd = ctr[d];
  float dx = cs.x - cd.x;
  float dy = cs.y - cd.y;
  if (fabsf(dx) < THR_X && fabsf(dy) < THR_Y) {
    float inv = FORCE_RATIO / (dx * dx + dy * dy + EPS);
    float fx = dx * inv;
    float fy = dy * inv;
    unsafeAtomicAdd(&frc[s].x,  fx);
    unsafeAtomicAdd(&frc[s].y,  fy);
    unsafeAtomicAdd(&frc[d].x, -fx);
    unsafeAtomicAdd(&frc[d].y, -fy);
  }
}

__global__ void repulsion_pairs(const int* __restrict__ src,
                                const int* __restrict__ dst,
                                const float2* __restrict__ ctr,
                                float2* __restrict__ frc, int e) {
  int i = blockIdx.x * blockDim.x + threadIdx.x;
  int base = i * 4;
  if (base + 3 < e) {
    i4v s4 = __builtin_nontemporal_load((const i4v*)(src + base));
    i4v d4 = __builtin_nontemporal_load((const i4v*)(dst + base));
    do_pair(s4.x, d4.x, ctr, frc);
    do_pair(s4.y, d4.y, ctr, frc);
    do_pair(s4.z, d4.z, ctr, frc);
    do_pair(s4.w, d4.w, ctr, frc);
  } else {
    for (int k = base; k < e; ++k) do_pair(src[k], dst[k], ctr, frc);
  }
}

// ---------------------------------------------------------------------------
// Kernel 3: abs-sum reduction with a WMMA wave32 reduction.
// A (16x4 f32) holds each lane's partial exactly once (VGPR0 = s, VGPR1 = 0:
// lanes 0-15 -> A[m][0], lanes 16-31 -> A[m][2]).  B = ones.
// D[m][n] = s_m + s_{m+16}  (independent of n).
// Lane 0's 8 D-VGPRs sum rows 0-7; lane 16's sum rows 8-15.
// ---------------------------------------------------------------------------
__global__ void repulsion_reduce(const float2* __restrict__ frc,
                                 float* __restrict__ out, int n) {
  int tid = blockIdx.x * blockDim.x + threadIdx.x;
  int stride = gridDim.x * blockDim.x;
  float s = 0.0f;
  const v2f* f2 = (const v2f*)frc;
  for (int i = tid; i < n; i += stride) {
    v2f f = __builtin_nontemporal_load(f2 + i);   // read-once stream
    s += __builtin_fabsf(f.x) + __builtin_fabsf(f.y);
  }

  v2f a; a.x = s;    a.y = 0.0f;
  v2f b; b.x = 1.0f; b.y = 1.0f;
  v8f c = {};
  c = __builtin_amdgcn_wmma_f32_16x16x4_f32(false, a, false, b,
                                            (short)0, c, false, false);
  float t = c[0] + c[1] + c[2] + c[3] + c[4] + c[5] + c[6] + c[7];
  t += __shfl_xor(t, 16, 32);            // lane0 now has full 32-lane sum
  if ((threadIdx.x & 31) == 0) unsafeAtomicAdd(out, t);
}

// ---------------------------------------------------------------------------
extern "C" void kernel_launch(void* const* d_in, const int* in_sizes, int n_in,
                              void* d_out, int out_size, void* d_ws, size_t ws_size,
                              hipStream_t stream) {
  const float* pos = (const float*)d_in[0];
  const float* nsx = (const float*)d_in[1];
  const float* nsy = (const float*)d_in[2];
  const int*   src = (const int*)d_in[3];
  const int*   dst = (const int*)d_in[4];
  float* out = (float*)d_out;

  int n = in_sizes[1];   // node count N
  int e = in_sizes[3];   // pair count E

  float2* ctr = (float2*)d_ws;
  float2* frc = ctr + n;

  const int B = 256;
  int gi = (n + B - 1) / B;
  repulsion_init<<<gi, B, 0, stream>>>(pos, nsx, nsy, ctr, frc, out, n);

  int e4 = (e + 3) / 4;
  int gp = (e4 + B - 1) / B;
  repulsion_pairs<<<gp, B, 0, stream>>>(src, dst, ctr, frc, e);

  int gr = 2048;  // grid-stride; full blocks keep EXEC all-ones for WMMA
  repulsion_reduce<<<gr, B, 0, stream>>>(frc, out, n);
}